// HeteroGNN_89429809037886
// MI455X (gfx1250) — compile-verified
//
#include <hip/hip_runtime.h>

typedef __attribute__((ext_vector_type(2))) float v2f;
typedef __attribute__((ext_vector_type(8))) float v8f;

#define NN    50000
#define EE    800000
#define HH    256
#define HHALF 128
#define OUTC  16
#define KSAGE 768          // [mean0 | mean1 | h] concatenated K
#define SLOPE 0.1f
#define BNEPS 1e-5f
#define MTILES (NN / 16)   // 3125

__device__ __forceinline__ float leaky(float v) { return v > 0.f ? v : SLOPE * v; }

// ---------------------------------------------------------------- utilities
__global__ void zero_kernel(float4* __restrict__ p, long n4) {
    long i = (long)blockIdx.x * blockDim.x + threadIdx.x;
    long stride = (long)gridDim.x * blockDim.x;
    float4 z = make_float4(0.f, 0.f, 0.f, 0.f);
    for (; i < n4; i += stride) p[i] = z;
}

__global__ void count_kernel(const int* __restrict__ dst, float* __restrict__ cnt, int e) {
    int i = blockIdx.x * blockDim.x + threadIdx.x;
    if (i < e) unsafeAtomicAdd(&cnt[dst[i]], 1.0f);
}

__global__ void inv_kernel(const float* __restrict__ cnt, float* __restrict__ inv, int n) {
    int i = blockIdx.x * blockDim.x + threadIdx.x;
    if (i < n) inv[i] = 1.0f / fmaxf(cnt[i], 1.0f);
}

// Wcomb[k][n]: k<256 -> Wl[l,0]; k<512 -> Wl[l,1]; else Wr[l,0]+Wr[l,1]. badd = bl[l,0]+bl[l,1].
__global__ void prepack_kernel(const float* __restrict__ Wl, const float* __restrict__ bl,
                               const float* __restrict__ Wr, float* __restrict__ Wc,
                               float* __restrict__ badd, int l) {
    int i = blockIdx.x * blockDim.x + threadIdx.x;
    if (i < KSAGE * HH) {
        int k = i / HH, n = i % HH;
        float v;
        if (k < 256)      v = Wl[((l * 2 + 0) * HH + k) * HH + n];
        else if (k < 512) v = Wl[((l * 2 + 1) * HH + (k - 256)) * HH + n];
        else              v = Wr[((l * 2 + 0) * HH + (k - 512)) * HH + n]
                            + Wr[((l * 2 + 1) * HH + (k - 512)) * HH + n];
        Wc[i] = v;
    }
    if (i < HH) badd[i] = bl[(l * 2 + 0) * HH + i] + bl[(l * 2 + 1) * HH + i];
}

// ------------------------------------------------------- edge scatter (mean numerator)
__global__ void aggregate_kernel(const float* __restrict__ h, const int* __restrict__ src,
                                 const int* __restrict__ dst, float* __restrict__ agg) {
    long total = (long)EE * 64;               // float4 chunks per edge
    long i = (long)blockIdx.x * blockDim.x + threadIdx.x;
    long stride = (long)gridDim.x * blockDim.x;
    for (; i < total; i += stride) {
        int e  = (int)(i >> 6);
        int c4 = (int)(i & 63);
        int s = src[e], d = dst[e];
        float4 v = ((const float4*)h)[(long)s * 64 + c4];
        float* o = agg + (long)d * HH + c4 * 4;
        unsafeAtomicAdd(o + 0, v.x);
        unsafeAtomicAdd(o + 1, v.y);
        unsafeAtomicAdd(o + 2, v.z);
        unsafeAtomicAdd(o + 3, v.w);
    }
}

// -------------------------------------------- async B-panel staging: global -> LDS
// Copies B[k, n0:n0+16] for k in [0,K) into Bs[k*16 ..] using the CDNA5 async-copy
// path (GLOBAL_LOAD_ASYNC_TO_LDS_B128, tracked by ASYNCcnt). 16 bytes per lane.
__device__ __forceinline__ void load_B_panel_async(const float* __restrict__ Bg, int K,
                                                   int Nc, int n0, float* Bs) {
    int q  = threadIdx.x & 3;    // 16-byte quarter of a 16-float row
    int kt = threadIdx.x >> 2;   // 0..63: k-row within the 64-row slab
    for (int k0 = 0; k0 < K; k0 += 64) {
        int k = k0 + kt;
        unsigned lds_addr = (unsigned)(uintptr_t)&Bs[(size_t)k * 16 + q * 4];
        unsigned long long gaddr =
            (unsigned long long)(uintptr_t)(Bg + (size_t)k * Nc + n0 + q * 4);
        asm volatile("global_load_async_to_lds_b128 %0, %1, off"
                     :: "v"(lds_addr), "v"(gaddr) : "memory");
    }
    asm volatile("s_wait_asynccnt 0x0" ::: "memory");
    __syncthreads();
}

// ------------------------------------------------------- WMMA f32 16x16x4 building block
// B fragment comes from the LDS panel (16 floats per k-row).
__device__ __forceinline__ v8f wmma_k256_lds(const float* __restrict__ arow, float scale,
                                             const float* Bs, int kbase,
                                             int row, int kh, v8f acc) {
#pragma unroll 8
    for (int kk = 0; kk < 256; kk += 4) {
        int k2 = kk + 2 * kh;                       // lane-half K offset
        v2f a, b;
        a.x = arow[k2] * scale;
        a.y = arow[k2 + 1] * scale;
        b.x = Bs[(size_t)(kbase + k2) * 16 + row];
        b.y = Bs[(size_t)(kbase + k2 + 1) * 16 + row];
        acc = __builtin_amdgcn_wmma_f32_16x16x4_f32(false, a, false, b, (short)0, acc,
                                                    false, false);
    }
    return acc;
}

// out = leaky( [mean0|mean1|h] @ Wcomb + badd ), one 16x16 tile per wave,
// 8 waves/block share one async-staged B panel (768x16 = 48KB LDS).
__global__ void sage_gemm_kernel(const float* __restrict__ agg0, const float* __restrict__ agg1,
                                 const float* __restrict__ hin, const float* __restrict__ inv0,
                                 const float* __restrict__ inv1, const float* __restrict__ W,
                                 const float* __restrict__ bias, float* __restrict__ out) {
    __shared__ float Bs[KSAGE * 16];
    int n0 = blockIdx.y * 16;
    load_B_panel_async(W, KSAGE, HH, n0, Bs);

    int wave = threadIdx.x >> 5;
    int lane = threadIdx.x & 31;
    int tm = blockIdx.x * 8 + wave;
    if (tm >= MTILES) return;                      // wave-uniform: EXEC stays all-ones
    int row = lane & 15;
    int kh  = lane >> 4;
    int m0 = tm * 16;
    int ncol = n0 + row;
    size_t arow = (size_t)(m0 + row) * HH;
    float s0 = inv0[m0 + row];
    float s1 = inv1[m0 + row];

    v8f acc = {};
    acc = wmma_k256_lds(agg0 + arow, s0,   Bs,   0, row, kh, acc);
    acc = wmma_k256_lds(agg1 + arow, s1,   Bs, 256, row, kh, acc);
    acc = wmma_k256_lds(hin  + arow, 1.0f, Bs, 512, row, kh, acc);

#pragma unroll
    for (int r = 0; r < 8; ++r) {
        int m = m0 + r + 8 * kh;                   // C layout: vgpr r -> M=r / M=r+8
        out[(size_t)m * HH + ncol] = leaky(acc[r] + bias[ncol]);
    }
}

// generic C[M x Nc] = act(A[M x K] @ B[K x Nc] + bias), async-staged B panel
__global__ void mlp_gemm_kernel(const float* __restrict__ A, const float* __restrict__ B,
                                const float* __restrict__ bias, float* __restrict__ C,
                                int K, int Nc, int do_leaky) {
    __shared__ float Bs[HH * 16];
    int n0 = blockIdx.y * 16;
    load_B_panel_async(B, K, Nc, n0, Bs);

    int wave = threadIdx.x >> 5;
    int lane = threadIdx.x & 31;
    int tm = blockIdx.x * 8 + wave;
    if (tm >= MTILES) return;
    int row = lane & 15;
    int kh  = lane >> 4;
    int m0 = tm * 16;
    int ncol = n0 + row;
    const float* arow = A + (size_t)(m0 + row) * K;

    v8f acc = {};
#pragma unroll 8
    for (int kk = 0; kk < K; kk += 4) {
        int k2 = kk + 2 * kh;
        v2f a, b;
        a.x = arow[k2];
        a.y = arow[k2 + 1];
        b.x = Bs[(size_t)k2 * 16 + row];
        b.y = Bs[(size_t)(k2 + 1) * 16 + row];
        acc = __builtin_amdgcn_wmma_f32_16x16x4_f32(false, a, false, b, (short)0, acc,
                                                    false, false);
    }
#pragma unroll
    for (int r = 0; r < 8; ++r) {
        int m = m0 + r + 8 * kh;
        float v = acc[r] + bias[ncol];
        C[(size_t)m * Nc + ncol] = do_leaky ? leaky(v) : v;
    }
}

// ------------------------------------------------------- BatchNorm
__global__ void bn_reduce_kernel(const float* __restrict__ hn, float* __restrict__ sum,
                                 float* __restrict__ sumsq) {
    int t = threadIdx.x;                           // column 0..255
    int r0 = blockIdx.x * 128;
    int r1 = r0 + 128 < NN ? r0 + 128 : NN;
    float s = 0.f, q = 0.f;
    for (int r = r0; r < r1; ++r) {
        float v = hn[(size_t)r * HH + t];
        s += v; q += v * v;
    }
    unsafeAtomicAdd(&sum[t], s);
    unsafeAtomicAdd(&sumsq[t], q);
}

__global__ void bn_norm_kernel(const float* __restrict__ hn, const float* __restrict__ sum,
                               const float* __restrict__ sumsq, const float* __restrict__ gamma,
                               const float* __restrict__ beta, float* __restrict__ out) {
    long total = (long)NN * HH;
    long i = (long)blockIdx.x * blockDim.x + threadIdx.x;
    long stride = (long)gridDim.x * blockDim.x;
    const float invN = 1.0f / (float)NN;
    for (; i < total; i += stride) {
        int c = (int)(i & (HH - 1));
        float mu  = sum[c] * invN;
        float var = sumsq[c] * invN - mu * mu;
        float sc  = gamma[c] * rsqrtf(var + BNEPS);
        out[i] = (hn[i] - mu) * sc + beta[c];
    }
}

// ---------------------------------------------------------------- driver
extern "C" void kernel_launch(void* const* d_in, const int* in_sizes, int n_in,
                              void* d_out, int out_size, void* d_ws, size_t ws_size,
                              hipStream_t stream) {
    const float* x    = (const float*)d_in[0];
    const int* src0   = (const int*)d_in[1];
    const int* dst0   = (const int*)d_in[2];
    const int* src1   = (const int*)d_in[3];
    const int* dst1   = (const int*)d_in[4];
    const float* Wl   = (const float*)d_in[5];
    const float* bl   = (const float*)d_in[6];
    const float* Wr   = (const float*)d_in[7];
    const float* gamma= (const float*)d_in[8];
    const float* beta = (const float*)d_in[9];
    const float* W1   = (const float*)d_in[10];
    const float* b1   = (const float*)d_in[11];
    const float* W2   = (const float*)d_in[12];
    const float* b2   = (const float*)d_in[13];
    float* out = (float*)d_out;

    float* ws = (float*)d_ws;
    size_t off = 0;
    float* agg0 = ws + off; off += (size_t)NN * HH;
    float* agg1 = ws + off; off += (size_t)NN * HH;
    float* hn   = ws + off; off += (size_t)NN * HH;
    float* hcur = ws + off; off += (size_t)NN * HH;
    float* z1   = ws + off; off += (size_t)NN * HHALF;
    float* Wc   = ws + off; off += (size_t)KSAGE * HH;
    float* badd = ws + off; off += HH;
    float* inv0 = ws + off; off += NN;
    float* inv1 = ws + off; off += NN;
    float* cnt0 = ws + off; off += NN;
    float* cnt1 = ws + off; off += NN;
    float* bsum = ws + off; off += HH;
    float* bsq  = ws + off; off += HH;

    const int T = 256;
    // in-degree counts (edge lists constant across layers)
    zero_kernel<<<256, T, 0, stream>>>((float4*)cnt0, NN / 4);
    zero_kernel<<<256, T, 0, stream>>>((float4*)cnt1, NN / 4);
    count_kernel<<<(EE + T - 1) / T, T, 0, stream>>>(dst0, cnt0, EE);
    count_kernel<<<(EE + T - 1) / T, T, 0, stream>>>(dst1, cnt1, EE);
    inv_kernel<<<(NN + T - 1) / T, T, 0, stream>>>(cnt0, inv0, NN);
    inv_kernel<<<(NN + T - 1) / T, T, 0, stream>>>(cnt1, inv1, NN);

    dim3 ggemm((MTILES + 7) / 8, HH / 16);
    const float* h = x;
    for (int l = 0; l < 2; ++l) {
        prepack_kernel<<<(KSAGE * HH + T - 1) / T, T, 0, stream>>>(Wl, bl, Wr, Wc, badd, l);
        zero_kernel<<<4096, T, 0, stream>>>((float4*)agg0, (long)NN * HH / 4);
        zero_kernel<<<4096, T, 0, stream>>>((float4*)agg1, (long)NN * HH / 4);
        aggregate_kernel<<<8192, T, 0, stream>>>(h, src0, dst0, agg0);
        aggregate_kernel<<<8192, T, 0, stream>>>(h, src1, dst1, agg1);
        sage_gemm_kernel<<<ggemm, T, 0, stream>>>(agg0, agg1, h, inv0, inv1, Wc, badd, hn);
        zero_kernel<<<1, 128, 0, stream>>>((float4*)bsum, HH / 4);
        zero_kernel<<<1, 128, 0, stream>>>((float4*)bsq, HH / 4);
        bn_reduce_kernel<<<(NN + 127) / 128, T, 0, stream>>>(hn, bsum, bsq);
        bn_norm_kernel<<<4096, T, 0, stream>>>(hn, bsum, bsq, gamma + l * HH, beta + l * HH, hcur);
        h = hcur;
    }
    dim3 g1((MTILES + 7) / 8, HHALF / 16);
    mlp_gemm_kernel<<<g1, T, 0, stream>>>(hcur, W1, b1, z1, HH, HHALF, 1);
    dim3 g2((MTILES + 7) / 8, OUTC / 16);
    mlp_gemm_kernel<<<g2, T, 0, stream>>>(z1, W2, b2, out, HHALF, OUTC, 0);
}